// ParallelBackProjection2D_50422916055306
// MI455X (gfx1250) — compile-verified
//
#include <hip/hip_runtime.h>
#include <math.h>

// Problem constants (fixed by the reference).
#define B_  4
#define A_  512
#define D_  729
#define H_  512
#define W_  512
#define TILE 32          // pixel tile is TILE x TILE
#define WIN  64          // detector window per (tile, angle); span <= 31*sqrt(2)+3 < 64
#define NT   256         // threads per block (8 wave32)

__global__ __launch_bounds__(NT)
void ParallelBackProjection2D_kernel(const float* __restrict__ sino,
                                     const float* __restrict__ vol_origin,
                                     const float* __restrict__ det_origin,
                                     const float* __restrict__ vol_spacing,
                                     const float* __restrict__ det_spacing,
                                     const float* __restrict__ angles,
                                     float* __restrict__ out)
{
    __shared__ float s_cos[A_];
    __shared__ float s_sin[A_];
    // Double-buffered window, transposed layout: [slot][bin][batch] -> float4 per bin.
    __shared__ __align__(16) float s_buf[2][WIN * B_];

    const int tid = threadIdx.x;

    // ---- Phase 0: trig table + zero buffers ----------------------------------
    for (int a = tid; a < A_; a += NT) {
        float th = angles[a];
        s_cos[a] = cosf(th);
        s_sin[a] = sinf(th);
    }
    for (int i = tid; i < 2 * WIN * B_; i += NT) {
        ((float*)s_buf)[i] = 0.0f;
    }

    const float voy  = vol_origin[0];
    const float vox  = vol_origin[1];
    const float spy  = vol_spacing[0];
    const float spx  = vol_spacing[1];
    const float det0 = det_origin[0];
    const float dinv = 1.0f / det_spacing[0];

    __syncthreads();

    // ---- Per-thread pixel assignment -----------------------------------------
    const int x0  = blockIdx.x * TILE;
    const int y0  = blockIdx.y * TILE;
    const int tx  = tid & (TILE - 1);   // 0..31
    const int tyb = tid >> 5;           // 0..7 ; rows tyb, tyb+8, tyb+16, tyb+24

    const float xf = vox + (float)(x0 + tx) * spx;
    float yf[4];
#pragma unroll
    for (int k = 0; k < 4; ++k)
        yf[k] = voy + (float)(y0 + tyb + 8 * k) * spy;

    // Tile corner coords (uniform) for the window base.
    const float cx0 = vox + (float)x0 * spx;
    const float cx1 = vox + (float)(x0 + TILE - 1) * spx;
    const float cy0 = voy + (float)y0 * spy;
    const float cy1 = voy + (float)(y0 + TILE - 1) * spy;

    float acc[4][4];
#pragma unroll
    for (int k = 0; k < 4; ++k)
#pragma unroll
        for (int b = 0; b < 4; ++b) acc[k][b] = 0.0f;

    // Staging assignment: thread loads detector bin (w0+lj) of batch lb.
    const int lb = tid >> 6;         // 0..3
    const int lj = tid & (WIN - 1);  // 0..63

    auto window_base = [&](int a) -> int {
        float c = s_cos[a], s = s_sin[a];
        float s00 = c * cx0 + s * cy0;
        float s01 = c * cx0 + s * cy1;
        float s10 = c * cx1 + s * cy0;
        float s11 = c * cx1 + s * cy1;
        float smin = fminf(fminf(s00, s01), fminf(s10, s11));
        return (int)floorf((smin - det0) * dinv) - 1;
    };

    auto issue = [&](int a, int slot, int w0) {
        int d = w0 + lj;
        if (d >= 0 && d < D_) {
            // LDS byte offset of s_buf[slot][lj*4 + lb] (low 32 bits of generic ptr).
            unsigned loff = (unsigned)(size_t)(&s_buf[slot][lj * 4 + lb]);
            unsigned long long ga =
                (unsigned long long)(size_t)(sino + (((size_t)lb * A_ + (size_t)a) * D_ + (size_t)d));
            // CDNA5 async global->LDS copy (tracked by ASYNCcnt).
            asm volatile("global_load_async_to_lds_b32 %0, %1, off"
                         :: "v"(loff), "v"(ga)
                         : "memory");
        }
    };

    // ---- Prefetch angle 0 -----------------------------------------------------
    int w_next = window_base(0);
    issue(0, 0, w_next);
    asm volatile("s_wait_asynccnt 0x0" ::: "memory");
    __syncthreads();

    int w_cur = w_next;

    // ---- Main angle loop (double buffered) -----------------------------------
    for (int a = 0; a < A_; ++a) {
        const int slot = a & 1;

        if (a + 1 < A_) {
            w_next = window_base(a + 1);
            issue(a + 1, slot ^ 1, w_next);   // other slot's readers passed last barrier
        }

        const float c  = s_cos[a];
        const float s  = s_sin[a];
        const float sx = xf * c;
        const float4* bufv = reinterpret_cast<const float4*>(&s_buf[slot][0]);

#pragma unroll
        for (int k = 0; k < 4; ++k) {
            float u  = (sx + yf[k] * s - det0) * dinv;
            float u0 = floorf(u);
            float fr = u - u0;
            int   i0 = (int)u0;
            int   i1 = i0 + 1;
            bool  v0 = (i0 >= 0) && (i0 < D_);
            bool  v1 = (i1 >= 0) && (i1 < D_);
            float w0v = v0 ? (1.0f - fr) : 0.0f;
            float w1v = v1 ? fr : 0.0f;
            int j0 = i0 - w_cur;
            if (j0 < 0) j0 = 0;
            if (j0 > WIN - 2) j0 = WIN - 2;
            float4 p0 = bufv[j0];
            float4 p1 = bufv[j0 + 1];
            acc[k][0] = fmaf(w0v, p0.x, fmaf(w1v, p1.x, acc[k][0]));
            acc[k][1] = fmaf(w0v, p0.y, fmaf(w1v, p1.y, acc[k][1]));
            acc[k][2] = fmaf(w0v, p0.z, fmaf(w1v, p1.z, acc[k][2]));
            acc[k][3] = fmaf(w0v, p0.w, fmaf(w1v, p1.w, acc[k][3]));
        }

        // Next window must be resident before the buffer swap; barrier also
        // protects the WAR on slot^1 for the following iteration's issue.
        asm volatile("s_wait_asynccnt 0x0" ::: "memory");
        __syncthreads();
        w_cur = w_next;
    }

    // ---- Write out ------------------------------------------------------------
#pragma unroll
    for (int k = 0; k < 4; ++k) {
        int y = y0 + tyb + 8 * k;
        int x = x0 + tx;
#pragma unroll
        for (int b = 0; b < 4; ++b)
            out[((size_t)b * H_ + (size_t)y) * W_ + (size_t)x] = acc[k][b];
    }
}

extern "C" void kernel_launch(void* const* d_in, const int* in_sizes, int n_in,
                              void* d_out, int out_size, void* d_ws, size_t ws_size,
                              hipStream_t stream) {
    (void)in_sizes; (void)n_in; (void)d_ws; (void)ws_size; (void)out_size;
    const float* sino        = (const float*)d_in[0];
    // d_in[1] = volume_shape (int64[2]) -- fixed at 512x512, unused.
    const float* vol_origin  = (const float*)d_in[2];
    const float* det_origin  = (const float*)d_in[3];
    const float* vol_spacing = (const float*)d_in[4];
    const float* det_spacing = (const float*)d_in[5];
    const float* angles      = (const float*)d_in[6];
    float* out               = (float*)d_out;

    dim3 grid(W_ / TILE, H_ / TILE);   // 16 x 16 tiles
    dim3 block(NT);
    ParallelBackProjection2D_kernel<<<grid, block, 0, stream>>>(
        sino, vol_origin, det_origin, vol_spacing, det_spacing, angles, out);
}